// Convolution_v1_13099650253153
// MI455X (gfx1250) — compile-verified
//
#include <hip/hip_runtime.h>
#include <math.h>

typedef __attribute__((ext_vector_type(2))) float v2f;
typedef __attribute__((ext_vector_type(8))) float v8f;

#define B_DIM 2
#define N_NODES 25000
#define FC_IN 64
#define FC_HID 256
#define MUL 32
#define OUT_CH 288
#define EDGES_PER_BLOCK 128
#define WAVES 8
#define SH_STRIDE 18   // per-wave H tile row stride (16 + 2 pad, even for b64 LDS loads)
#define SW_STRIDE 97   // w staging row stride (96 + 1 pad)

// Native no-return FP32 atomic add (GLOBAL_ATOMIC_ADD_F32). unsafeAtomicAdd is the
// HIP API guaranteed to emit the hardware FP atomic on AMD targets.
__device__ __forceinline__ void atomic_add_f32(float* p, float v) {
    unsafeAtomicAdd(p, v);
}

__global__ __launch_bounds__(256) void conv_mlp_tp_scatter(
    const int* __restrict__ edge_src,
    const int* __restrict__ edge_dst,
    const float* __restrict__ node_emb,
    const float* __restrict__ edge_type,
    const float* __restrict__ W1,
    const float* __restrict__ W2,
    float* __restrict__ out)
{
    __shared__ float sH[WAVES * 16 * SH_STRIDE];      // per-wave C->A relayout tile
    __shared__ float sW[EDGES_PER_BLOCK * SW_STRIDE]; // per-block w = MLP(edge_type)

    const int tid   = threadIdx.x;
    const int wv    = tid >> 5;
    const int lane  = tid & 31;
    const int lmod  = lane & 15;
    const int khalf = (lane < 16) ? 0 : 2;      // K sub-offset for 16x16x4 A/B frags
    const int e_base = blockIdx.x * EDGES_PER_BLOCK;
    const int e_tile = e_base + wv * 16;        // this wave's 16-edge row tile

    const v8f vzero = {0.f, 0.f, 0.f, 0.f, 0.f, 0.f, 0.f, 0.f};

    // ---------------- Phase 1: A-fragments of edge_type tile (16 x 64) ----------------
    // A layout (32-bit 16x4): lane L holds M=L%16; vgpr0=K=(khalf), vgpr1=K=(khalf+1)
    v2f aET[16];
    {
        const float* etp = edge_type + (size_t)(e_tile + lmod) * FC_IN;
        #pragma unroll
        for (int ks = 0; ks < 16; ++ks)
            aET[ks] = *(const v2f*)(etp + 4 * ks + khalf);
        // Prefetch next block's edge_type tile rows into cache (gfx1250 global_prefetch_b8).
        __builtin_prefetch(etp + EDGES_PER_BLOCK * FC_IN, 0, 1);
    }

    // ---------------- GEMM1: H = silu((ET @ W1) * 1/sqrt(64)), 16 x 256 ----------------
    v8f h[16];
    #pragma unroll
    for (int nt = 0; nt < 16; ++nt) {
        v8f acc = vzero;
        #pragma unroll
        for (int ks = 0; ks < 16; ++ks) {
            const int k0 = 4 * ks + khalf;
            v2f b;  // B layout (4x16): vgpr0=B[khalf][n], vgpr1=B[khalf+1][n], n=L%16
            b.x = W1[(size_t)k0 * FC_HID + 16 * nt + lmod];
            b.y = W1[(size_t)(k0 + 1) * FC_HID + 16 * nt + lmod];
            acc = __builtin_amdgcn_wmma_f32_16x16x4_f32(
                false, aET[ks], false, b, (short)0, acc, false, false);
        }
        #pragma unroll
        for (int i = 0; i < 8; ++i) {
            const float hv = acc[i] * 0.125f;          // 1/sqrt(64)
            acc[i] = hv / (1.0f + __expf(-hv));        // silu
        }
        h[nt] = acc;
    }

    // ---------------- GEMM2: w = (H @ W2) * 1/sqrt(256), 16 x 96 ----------------
    float* myH = sH + wv * (16 * SH_STRIDE);
    v8f acc2[6];
    #pragma unroll
    for (int j = 0; j < 6; ++j) acc2[j] = vzero;

    #pragma unroll
    for (int nt1 = 0; nt1 < 16; ++nt1) {
        // Spill C-layout tile h[nt1] to LDS row-major (C layout: vgpr r, lane L ->
        // M = r + (L<16?0:8), N = L%16), then reload as A-fragments.
        {
            const int mbase = (lane < 16) ? 0 : 8;
            #pragma unroll
            for (int r = 0; r < 8; ++r)
                myH[(r + mbase) * SH_STRIDE + lmod] = h[nt1][r];
        }
        v2f a[4];
        #pragma unroll
        for (int kk = 0; kk < 4; ++kk)
            a[kk] = *(const v2f*)(myH + lmod * SH_STRIDE + 4 * kk + khalf);

        #pragma unroll
        for (int nt2 = 0; nt2 < 6; ++nt2) {
            #pragma unroll
            for (int kk = 0; kk < 4; ++kk) {
                const int k0 = 16 * nt1 + 4 * kk + khalf;
                v2f b;
                b.x = W2[(size_t)k0 * 96 + 16 * nt2 + lmod];
                b.y = W2[(size_t)(k0 + 1) * 96 + 16 * nt2 + lmod];
                acc2[nt2] = __builtin_amdgcn_wmma_f32_16x16x4_f32(
                    false, a[kk], false, b, (short)0, acc2[nt2], false, false);
            }
        }
    }

    // Stage w tiles into LDS row-major [128][SW_STRIDE], scaled by 1/sqrt(256)
    {
        const int mbase = wv * 16 + ((lane < 16) ? 0 : 8);
        #pragma unroll
        for (int nt2 = 0; nt2 < 6; ++nt2) {
            #pragma unroll
            for (int r = 0; r < 8; ++r)
                sW[(mbase + r) * SW_STRIDE + 16 * nt2 + lmod] = acc2[nt2][r] * 0.0625f;
        }
    }
    __syncthreads();

    // ---------------- Phase 3: tensor product + scatter-add (atomics into L2) --------
    {
        const int e_loc = tid >> 1;      // 0..127
        const int bb    = tid & 1;       // batch
        const int e     = e_base + e_loc;
        const int si    = edge_src[e];
        const int di    = edge_dst[e];
        const float* xp = node_emb + ((size_t)bb * N_NODES + si) * 3;
        const float* yp = node_emb + ((size_t)bb * N_NODES + di) * 3;
        const float x0 = xp[0], x1 = xp[1], x2 = xp[2];
        const float y0 = yp[0], y1 = yp[1], y2 = yp[2];

        const float inv3 = 0.57735026918962576f;  // 1/sqrt(3)
        const float inv2 = 0.70710678118654752f;  // 1/sqrt(2)
        const float inv6 = 0.40824829046386302f;  // 1/sqrt(6)
        const float scl  = 0.25f;                 // 1/sqrt(NUM_NEIGHBORS)

        const float s  = (x0*y0 + x1*y1 + x2*y2) * inv3 * scl;
        const float v0 = (x1*y2 - x2*y1) * inv2 * scl;
        const float v1 = (x2*y0 - x0*y2) * inv2 * scl;
        const float v2 = (x0*y1 - x1*y0) * inv2 * scl;
        const float t0 = (x0*y1 + x1*y0) * inv2 * scl;
        const float t1 = (x1*y2 + x2*y1) * inv2 * scl;
        const float t2 = (x0*y2 + x2*y0) * inv2 * scl;
        const float t3 = (x0*y0 - x1*y1) * inv2 * scl;
        const float t4 = (2.0f*x2*y2 - x0*y0 - x1*y1) * inv6 * scl;

        const float* wrow = sW + e_loc * SW_STRIDE;
        float* op = out + ((size_t)bb * N_NODES + di) * OUT_CH;

        #pragma unroll 4
        for (int u = 0; u < MUL; ++u)
            atomic_add_f32(op + u, wrow[u] * s);

        #pragma unroll 4
        for (int u = 0; u < MUL; ++u) {
            const float wu = wrow[MUL + u];
            atomic_add_f32(op + MUL + 3*u + 0, wu * v0);
            atomic_add_f32(op + MUL + 3*u + 1, wu * v1);
            atomic_add_f32(op + MUL + 3*u + 2, wu * v2);
        }

        #pragma unroll 2
        for (int u = 0; u < MUL; ++u) {
            const float wu = wrow[2*MUL + u];
            atomic_add_f32(op + 4*MUL + 5*u + 0, wu * t0);
            atomic_add_f32(op + 4*MUL + 5*u + 1, wu * t1);
            atomic_add_f32(op + 4*MUL + 5*u + 2, wu * t2);
            atomic_add_f32(op + 4*MUL + 5*u + 3, wu * t3);
            atomic_add_f32(op + 4*MUL + 5*u + 4, wu * t4);
        }
    }
}

extern "C" void kernel_launch(void* const* d_in, const int* in_sizes, int n_in,
                              void* d_out, int out_size, void* d_ws, size_t ws_size,
                              hipStream_t stream) {
    const int*   edge_src  = (const int*)d_in[0];
    const int*   edge_dst  = (const int*)d_in[1];
    const float* node_emb  = (const float*)d_in[2];
    const float* edge_type = (const float*)d_in[3];
    const float* W1        = (const float*)d_in[4];
    const float* W2        = (const float*)d_in[5];
    float*       out       = (float*)d_out;

    // Harness poisons d_out; scatter needs zeros. Async memset is capture-safe.
    hipMemsetAsync(d_out, 0, (size_t)out_size * sizeof(float), stream);

    const int E = in_sizes[0];                    // 400000
    const int nblocks = E / EDGES_PER_BLOCK;      // 3125 exactly
    conv_mlp_tp_scatter<<<nblocks, 256, 0, stream>>>(
        edge_src, edge_dst, node_emb, edge_type, W1, W2, out);
}